// RoIPooling2D_9663676416511
// MI455X (gfx1250) — compile-verified
//
#include <hip/hip_runtime.h>
#include <stdint.h>

// RoI max-pool for MI455X (gfx1250, wave32).
// One wave per (roi, channel). Roi window rows are staged into a per-wave LDS
// slice with gfx1250 async global->LDS loads (ASYNCcnt domain), consumed from
// LDS, reduced across the wave with a 5-step shfl_xor butterfly.

#define C_CH   512
#define H_F    50
#define W_F    50
#define OUTH   7
#define CHROWS 4              // rows staged per async batch
#define SWPITCH 52            // padded row pitch in LDS (floats)
#define NEG_MAX (-3.402823466e38f)

__global__ __launch_bounds__(256) void roipool_kernel(
    const float* __restrict__ x, const float* __restrict__ rois,
    float* __restrict__ out, int total_waves)
{
  const int lane = threadIdx.x & 31;
  const int wib  = threadIdx.x >> 5;          // wave index inside block (0..7)
  const int wid  = (blockIdx.x << 3) + wib;   // global wave id
  if (wid >= total_waves) return;

  const int n = wid >> 9;                     // roi index   (wid / 512)
  const int c = wid & (C_CH - 1);             // channel     (wid % 512)

  // roi layout from the reference: [y1, x2, y2, x1]; /16 is exact, cast = trunc
  const float4 r4 = *(const float4*)(rois + 4 * n);
  const int y_lo = (int)(r4.x * 0.0625f);
  const int x_hi = (int)(r4.y * 0.0625f) + 1;   // exclusive
  const int y_hi = (int)(r4.z * 0.0625f) + 1;   // exclusive
  const int x_lo = (int)(r4.w * 0.0625f);

  const int span_h = y_hi - y_lo;               // 1..50
  const int span_w = x_hi - x_lo;               // 1..50

  // adaptive bins: start = y_lo + floor(b*h/7), end = y_lo + ceil((b+1)*h/7)
  int bs[OUTH], be[OUTH];
#pragma unroll
  for (int b = 0; b < OUTH; ++b) {
    bs[b] = y_lo + (b * span_h) / OUTH;
    be[b] = y_lo + ((b + 1) * span_h + OUTH - 1) / OUTH;
  }

  float acc[OUTH];
#pragma unroll
  for (int b = 0; b < OUTH; ++b) acc[b] = NEG_MAX;

  // private LDS slice per wave: 4 rows x 52 floats; 8 waves/block -> 6656 B
  __shared__ float smem[8 * CHROWS * SWPITCH];
  float* sw = &smem[wib * CHROWS * SWPITCH];

  const float* ch = x + (size_t)c * (H_F * W_F);

  for (int r0 = 0; r0 < span_h; r0 += CHROWS) {
    int rows = span_h - r0;
    if (rows > CHROWS) rows = CHROWS;

    // ---- stage: async global -> LDS (per-lane, EXEC-masked) ----
    for (int r = 0; r < rows; ++r) {
      const int y = y_lo + r0 + r;
      const float* rowp = ch + y * W_F + x_lo;
      if (lane < span_w) {
        uint32_t laddr = (uint32_t)(size_t)(&sw[r * SWPITCH + lane]);
        const float* g = rowp + lane;
        asm volatile("global_load_async_to_lds_b32 %0, %1, off"
                     :: "v"(laddr), "v"(g) : "memory");
      }
      if (lane + 32 < span_w) {
        uint32_t laddr = (uint32_t)(size_t)(&sw[r * SWPITCH + lane + 32]);
        const float* g = rowp + lane + 32;
        asm volatile("global_load_async_to_lds_b32 %0, %1, off"
                     :: "v"(laddr), "v"(g) : "memory");
      }
    }
    asm volatile("s_wait_asynccnt 0x0" ::: "memory");

    // ---- consume from LDS, update bin accumulators ----
    for (int r = 0; r < rows; ++r) {
      const int y = y_lo + r0 + r;
      float v = NEG_MAX;
      if (lane < span_w)      v = sw[r * SWPITCH + lane];
      if (lane + 32 < span_w) v = fmaxf(v, sw[r * SWPITCH + lane + 32]);
#pragma unroll
      for (int b = 0; b < OUTH; ++b)
        if (y >= bs[b] && y < be[b]) acc[b] = fmaxf(acc[b], v);
    }
  }

  // ---- wave32 butterfly max reduction ----
#pragma unroll
  for (int b = 0; b < OUTH; ++b) {
    float v = acc[b];
    v = fmaxf(v, __shfl_xor(v, 16, 32));
    v = fmaxf(v, __shfl_xor(v,  8, 32));
    v = fmaxf(v, __shfl_xor(v,  4, 32));
    v = fmaxf(v, __shfl_xor(v,  2, 32));
    v = fmaxf(v, __shfl_xor(v,  1, 32));
    acc[b] = v;
  }

  if (lane == 0) {
    float* o = out + ((size_t)n * C_CH + c) * OUTH;   // out[n, c*7 + b]
#pragma unroll
    for (int b = 0; b < OUTH; ++b) o[b] = acc[b];
  }
}

extern "C" void kernel_launch(void* const* d_in, const int* in_sizes, int n_in,
                              void* d_out, int out_size, void* d_ws, size_t ws_size,
                              hipStream_t stream) {
  const float* x    = (const float*)d_in[0];   // [1,512,50,50] f32
  const float* rois = (const float*)d_in[1];   // [N,4] f32, [y1,x2,y2,x1]
  float* out = (float*)d_out;                  // [N, 512*7] f32

  const int N = in_sizes[1] / 4;
  const int total_waves = N * C_CH;            // one wave per (roi, channel)
  const int blocks = (total_waves + 7) / 8;    // 8 waves per 256-thread block

  roipool_kernel<<<blocks, 256, 0, stream>>>(x, rois, out, total_waves);
}